// MultiheadAttention_35631048688005
// MI455X (gfx1250) — compile-verified
//
#include <hip/hip_runtime.h>
#include <math.h>

typedef float v2f __attribute__((ext_vector_type(2)));
typedef float v8f __attribute__((ext_vector_type(8)));
typedef unsigned int u32x4 __attribute__((ext_vector_type(4)));
typedef int i32x4 __attribute__((ext_vector_type(4)));
typedef int i32x8 __attribute__((ext_vector_type(8)));

static constexpr int Bb = 8, Nn = 128, Dd = 256, Hh = 8, DEe = 128, DGg = 256;

#if defined(__has_builtin)
#if __has_builtin(__builtin_amdgcn_tensor_load_to_lds)
#define USE_TDM 1
#endif
#endif

// ---------------------------------------------------------------------------
// fp32 WMMA GEMM, K=Nc=256 compile-time: C = (A @ W + bias) * scale * rowmask
// One 16x16 tile per wave. Each k16 group: preload 4 fragments, then 4 WMMAs,
// so loads cluster into one clause instead of draining loadcnt per WMMA.
// ---------------------------------------------------------------------------
__global__ void __launch_bounds__(256)
gemm_wmma_f32(const float* __restrict__ A, const float* __restrict__ W,
              const float* __restrict__ bias, const float* __restrict__ rowmask,
              float scale, float* __restrict__ C, int M) {
  constexpr int K = 256, Nc = 256;
  const int lane = threadIdx.x & 31;
  const int wave = threadIdx.x >> 5;
  constexpr int tilesN = Nc >> 4;
  const int tile = blockIdx.x * 8 + wave;
  const int m0 = (tile / tilesN) << 4;
  const int n0 = (tile % tilesN) << 4;
  const int l15 = lane & 15;
  const int kh = (lane < 16) ? 0 : 2;   // K-half held by upper lanes
  const int row = m0 + l15;
  const int col = n0 + l15;
  const float* __restrict__ Ar = A + row * K;
  const float* __restrict__ Wp = W + col;
  __builtin_prefetch(Wp, 0, 1);
  v8f acc = {};
  for (int k = 0; k < K; k += 16) {
    v2f a[4], b[4];
#pragma unroll
    for (int u = 0; u < 4; ++u) {
      const int kk = k + 4 * u + kh;
      a[u].x = Ar[kk];
      a[u].y = Ar[kk + 1];
      b[u].x = Wp[kk * Nc];
      b[u].y = Wp[(kk + 1) * Nc];
    }
#pragma unroll
    for (int u = 0; u < 4; ++u)
      acc = __builtin_amdgcn_wmma_f32_16x16x4_f32(false, a[u], false, b[u],
                                                  (short)0, acc, false, false);
  }
  const float bc = bias ? bias[col] : 0.0f;
#pragma unroll
  for (int r = 0; r < 8; ++r) {
    const int m = m0 + r + ((lane < 16) ? 0 : 8);
    float v = (acc[r] + bc) * scale;
    if (rowmask) v *= rowmask[m];
    C[m * Nc + col] = v;
  }
}

// ---------------------------------------------------------------------------
// Fused edge/attention kernel. One block per (b, t).
//  LDS: adj row-block [128x128], aw tile [128x256], q row, G1/G2 rows, bias col
//  adj staged via Tensor Data Mover (tensor_load_to_lds) when available.
// ---------------------------------------------------------------------------
__global__ void __launch_bounds__(256)
edge_attn_fused(const float* __restrict__ q, const float* __restrict__ kmat,
                const float* __restrict__ vmat, const float* __restrict__ adj,
                const float* __restrict__ We_mul, const float* __restrict__ be_mul,
                const float* __restrict__ We_add, const float* __restrict__ be_add,
                const float* __restrict__ We_out, const float* __restrict__ be_out,
                const float* __restrict__ G1, const float* __restrict__ G2,
                const float* __restrict__ attn_bias, const float* __restrict__ edge_mask,
                float* __restrict__ new_adj_out, float* __restrict__ attn_out) {
  extern __shared__ float smem[];
  float* s_adj  = smem;                       // 128*128
  float* s_aw   = s_adj + Nn * DEe;           // 128*256
  float* s_q    = s_aw + Nn * Dd;             // 256
  float* s_g1   = s_q + Dd;                   // 256
  float* s_g2   = s_g1 + Dd;                  // 256
  float* s_bias = s_g2 + Dd;                  // 128*8

  const int tid = threadIdx.x;
  const int lane = tid & 31;
  const int wave = tid >> 5;
  const int b = blockIdx.x >> 7;
  const int base_bt = blockIdx.x;             // b*N + t   (fits i32 everywhere)

  // ---- phase 0: stage operands into LDS ----
  const float* adj_bt = adj + base_bt * (Nn * DEe);
#ifdef USE_TDM
  // Tensor Data Mover: DMA the contiguous 128x128 f32 row-block into LDS.
  if (tid == 0) {
    const unsigned long long ga = (unsigned long long)(const void*)adj_bt;
    const unsigned lds_off = (unsigned)(unsigned long long)(void*)s_adj;
    constexpr unsigned ELEMS = Nn * DEe;      // 16384 4-byte elements
    u32x4 g0;
    g0.x = 1u;                                 // count=1, user descriptor
    g0.y = lds_off;                            // lds_addr (bytes)
    g0.z = (unsigned)(ga & 0xffffffffu);       // global_addr[31:0]
    g0.w = (unsigned)((ga >> 32) & 0x01ffffffu) | (2u << 30);  // [56:32] | type=2
    i32x8 g1;
    g1[0] = (int)(2u << 16);                   // wg_mask=0, data_size=2 (4 bytes)
    g1[1] = (int)((ELEMS & 0xffffu) << 16);    // tensor_dim0 lo16 at bits 63:48
    g1[2] = (int)((ELEMS >> 16) | (1u << 16)); // tensor_dim0 hi16 | tensor_dim1=1 lo16
    g1[3] = (int)((ELEMS & 0xffffu) << 16);    // tensor_dim1 hi16=0 | tile_dim0
    g1[4] = 1;                                 // tile_dim1=1, tile_dim2=0
    g1[5] = (int)ELEMS;                        // tensor_dim0_stride lo32
    g1[6] = 0;                                 // stride0 hi16 | stride1 lo16
    g1[7] = 0;                                 // stride1 hi32
    i32x4 gz = {0, 0, 0, 0};
    i32x8 gz8 = {0, 0, 0, 0, 0, 0, 0, 0};
    __builtin_amdgcn_tensor_load_to_lds(g0, g1, gz, gz, gz8, 0);
  }
  if (tid < 32) __builtin_amdgcn_s_wait_tensorcnt(0);
#else
  for (int i = tid; i < (Nn * DEe) / 4; i += 256)
    ((float4*)s_adj)[i] = ((const float4*)adj_bt)[i];
#endif
  for (int i = tid; i < Dd; i += 256) {
    s_q[i]  = q[base_bt * Dd + i];
    s_g1[i] = G1[b * Dd + i];
    s_g2[i] = G2[b * Dd + i];
  }
  for (int i = tid; i < Nn * Hh; i += 256)
    s_bias[i] = attn_bias[base_bt * (Nn * Hh) + i];
  __syncthreads();

  const int l15 = lane & 15;
  const int kh = (lane < 16) ? 0 : 2;

  // ---- phase 1: E1/E2 via WMMA (K=DE), fused into aw = q*k*(E1+1)+E2 ----
  for (int tile = wave; tile < 128; tile += 8) {   // 8 s-tiles x 16 d-tiles
    const int s0 = (tile >> 4) << 4;
    const int d0 = (tile & 15) << 4;
    const float* __restrict__ arow = s_adj + (s0 + l15) * DEe;
    const float* __restrict__ W1p = We_mul + d0 + l15;
    const float* __restrict__ W2p = We_add + d0 + l15;
    __builtin_prefetch(W1p, 0, 1);
    v8f e1 = {}, e2 = {};
    for (int k = 0; k < DEe; k += 16) {
      v2f a[4], b1[4], b2[4];
#pragma unroll
      for (int u = 0; u < 4; ++u) {
        const int kk = k + 4 * u + kh;
        a[u].x = arow[kk];
        a[u].y = arow[kk + 1];
        b1[u].x = W1p[kk * Dd];
        b1[u].y = W1p[(kk + 1) * Dd];
        b2[u].x = W2p[kk * Dd];
        b2[u].y = W2p[(kk + 1) * Dd];
      }
#pragma unroll
      for (int u = 0; u < 4; ++u) {
        e1 = __builtin_amdgcn_wmma_f32_16x16x4_f32(false, a[u], false, b1[u],
                                                   (short)0, e1, false, false);
        e2 = __builtin_amdgcn_wmma_f32_16x16x4_f32(false, a[u], false, b2[u],
                                                   (short)0, e2, false, false);
      }
    }
    const int d = d0 + l15;
    const float qd = s_q[d];
    const float bm = be_mul[d], ba = be_add[d];
#pragma unroll
    for (int r = 0; r < 8; ++r) {
      const int s = s0 + r + ((lane < 16) ? 0 : 8);
      const float em = edge_mask[base_bt * Nn + s];
      const float E1v = (e1[r] + bm) * em;
      const float E2v = (e2[r] + ba) * em;
      const float kv = kmat[(b * Nn + s) * Dd + d];
      s_aw[s * Dd + d] = qd * kv * (E1v + 1.0f) + E2v;
    }
  }
  __syncthreads();

  // ---- phase 2: per-channel softmax over src + attention reduction ----
  {
    const int d = tid;          // 256 threads -> one channel each
    const int h = d >> 5;       // Dh = 32
    float mx = -1e30f;
    for (int s = 0; s < Nn; ++s)
      mx = fmaxf(mx, s_aw[s * Dd + d] + s_bias[s * Hh + h]);
    float denom = 0.0f, acc = 0.0f;
    for (int s = 0; s < Nn; ++s) {
      const float p = __expf(s_aw[s * Dd + d] + s_bias[s * Hh + h] - mx);
      denom += p;
      acc += p * vmat[(b * Nn + s) * Dd + d];
    }
    attn_out[base_bt * Dd + d] = acc / denom;
  }
  __syncthreads();

  // ---- phase 3: in-place graph modulation of aw ----
  for (int i = tid; i < Nn * Dd; i += 256) {
    const int d = i & (Dd - 1);
    s_aw[i] = s_aw[i] * (s_g1[d] + 1.0f) + s_g2[d];
  }
  __syncthreads();

  // ---- phase 4: new_adj = aw_mod @ We_out (K=256) via WMMA ----
  for (int tile = wave; tile < 64; tile += 8) {    // 8 s-tiles x 8 e-tiles
    const int s0 = (tile >> 3) << 4;
    const int e0 = (tile & 7) << 4;
    const float* __restrict__ arow = s_aw + (s0 + l15) * Dd;
    const float* __restrict__ Wp = We_out + e0 + l15;
    __builtin_prefetch(Wp, 0, 1);
    v8f acc = {};
    for (int k = 0; k < Dd; k += 16) {
      v2f a[4], bf[4];
#pragma unroll
      for (int u = 0; u < 4; ++u) {
        const int kk = k + 4 * u + kh;
        a[u].x = arow[kk];
        a[u].y = arow[kk + 1];
        bf[u].x = Wp[kk * DEe];
        bf[u].y = Wp[(kk + 1) * DEe];
      }
#pragma unroll
      for (int u = 0; u < 4; ++u)
        acc = __builtin_amdgcn_wmma_f32_16x16x4_f32(false, a[u], false, bf[u],
                                                    (short)0, acc, false, false);
    }
    const int e = e0 + l15;
    const float be = be_out[e];
#pragma unroll
    for (int r = 0; r < 8; ++r) {
      const int s = s0 + r + ((lane < 16) ? 0 : 8);
      const float em = edge_mask[base_bt * Nn + s];
      new_adj_out[(base_bt * Nn + s) * DEe + e] = (acc[r] + be) * em;
    }
  }
}

// ---------------------------------------------------------------------------
// Node2Graph stats: mean/min/max/std (ddof=1) over N per (b,d)
// ---------------------------------------------------------------------------
__global__ void n2g_stats(const float* __restrict__ query, float* __restrict__ n2g_in) {
  const int idx = blockIdx.x * blockDim.x + threadIdx.x;   // 0..2047
  const int b = idx >> 8, d = idx & 255;
  float sum = 0.f, ss = 0.f, mn = 1e30f, mx = -1e30f;
  for (int s = 0; s < Nn; ++s) {
    const float x = query[(b * Nn + s) * Dd + d];
    sum += x; ss += x * x; mn = fminf(mn, x); mx = fmaxf(mx, x);
  }
  const float mean = sum / Nn;
  float var = (ss - Nn * mean * mean) / (Nn - 1);
  var = fmaxf(var, 0.0f);
  float* o = n2g_in + b * 4 * Dd;
  o[d] = mean; o[Dd + d] = mn; o[2 * Dd + d] = mx; o[3 * Dd + d] = sqrtf(var);
}

// ---------------------------------------------------------------------------
// Edge2Graph stats: over N*N per (b,e)
// ---------------------------------------------------------------------------
__global__ void e2g_stats(const float* __restrict__ adj, float* __restrict__ e2g_in) {
  const int idx = blockIdx.x * blockDim.x + threadIdx.x;   // 0..1023
  const int b = idx >> 7, e = idx & 127;
  const float* base = adj + b * (Nn * Nn * DEe) + e;
  float sum = 0.f, ss = 0.f, mn = 1e30f, mx = -1e30f;
  for (int i = 0; i < Nn * Nn; ++i) {
    const float x = base[i * DEe];
    sum += x; ss += x * x; mn = fminf(mn, x); mx = fmaxf(mx, x);
  }
  const float cnt = (float)(Nn * Nn);
  const float mean = sum / cnt;
  float var = (ss - cnt * mean * mean) / (cnt - 1.0f);
  var = fmaxf(var, 0.0f);
  float* o = e2g_in + b * 4 * DEe;
  o[e] = mean; o[DEe + e] = mn; o[2 * DEe + e] = mx; o[3 * DEe + e] = sqrtf(var);
}

// ---------------------------------------------------------------------------
// G1/G2 = graph_feature @ W_g_mul/add + bias   (B=8 rows; tiny)
// ---------------------------------------------------------------------------
__global__ void graph_mod(const float* __restrict__ gf,
                          const float* __restrict__ Wm, const float* __restrict__ bm,
                          const float* __restrict__ Wa, const float* __restrict__ ba,
                          float* __restrict__ G1, float* __restrict__ G2) {
  const int idx = blockIdx.x * blockDim.x + threadIdx.x;   // 0..2047
  const int b = idx >> 8, d = idx & 255;
  float s1 = bm[d], s2 = ba[d];
  for (int k = 0; k < DGg; ++k) {
    const float g = gf[b * DGg + k];
    s1 += g * Wm[k * Dd + d];
    s2 += g * Wa[k * Dd + d];
  }
  G1[b * Dd + d] = s1;
  G2[b * Dd + d] = s2;
}

// gsum = node2graph + gf + edge2graph
__global__ void graph_gather(const float* __restrict__ n2g_in,
                             const float* __restrict__ Wn, const float* __restrict__ bn,
                             const float* __restrict__ e2g_in,
                             const float* __restrict__ We, const float* __restrict__ be,
                             const float* __restrict__ gf, float* __restrict__ gsum) {
  const int idx = blockIdx.x * blockDim.x + threadIdx.x;   // 0..2047
  const int b = idx >> 8, d = idx & 255;
  float s = bn[d] + be[d] + gf[b * DGg + d];
  for (int k = 0; k < 4 * Dd; ++k) s += n2g_in[b * 4 * Dd + k] * Wn[k * DGg + d];
  for (int k = 0; k < 4 * DEe; ++k) s += e2g_in[b * 4 * DEe + k] * We[k * DGg + d];
  gsum[b * DGg + d] = s;
}

// new_graph_feature = gsum @ W_g_out + b_g_out
__global__ void graph_out_k(const float* __restrict__ gsum, const float* __restrict__ Wgo,
                            const float* __restrict__ bgo, float* __restrict__ out) {
  const int idx = blockIdx.x * blockDim.x + threadIdx.x;   // 0..2047
  const int b = idx >> 8, d = idx & 255;
  float s = bgo[d];
  for (int k = 0; k < DGg; ++k) s += gsum[b * DGg + k] * Wgo[k * DGg + d];
  out[b * DGg + d] = s;
}

// ---------------------------------------------------------------------------
extern "C" void kernel_launch(void* const* d_in, const int* in_sizes, int n_in,
                              void* d_out, int out_size, void* d_ws, size_t ws_size,
                              hipStream_t stream) {
  const float* query     = (const float*)d_in[0];
  const float* attn_bias = (const float*)d_in[3];
  const float* adj       = (const float*)d_in[4];
  const float* gfeat     = (const float*)d_in[5];
  const float* node_mask = (const float*)d_in[6];
  const float* edge_mask = (const float*)d_in[7];
  const float* W_q = (const float*)d_in[8],  *b_q = (const float*)d_in[9];
  const float* W_k = (const float*)d_in[10], *b_k = (const float*)d_in[11];
  const float* W_v = (const float*)d_in[12], *b_v = (const float*)d_in[13];
  const float* W_o = (const float*)d_in[14], *b_o = (const float*)d_in[15];
  const float* W_em = (const float*)d_in[16], *b_em = (const float*)d_in[17];
  const float* W_ea = (const float*)d_in[18], *b_ea = (const float*)d_in[19];
  const float* W_eo = (const float*)d_in[20], *b_eo = (const float*)d_in[21];
  const float* W_gm = (const float*)d_in[22], *b_gm = (const float*)d_in[23];
  const float* W_ga = (const float*)d_in[24], *b_ga = (const float*)d_in[25];
  const float* W_go = (const float*)d_in[26], *b_go = (const float*)d_in[27];
  const float* W_n2g = (const float*)d_in[28], *b_n2g = (const float*)d_in[29];
  const float* W_e2g = (const float*)d_in[30], *b_e2g = (const float*)d_in[31];

  float* out = (float*)d_out;                         // [B,N,D] = 262144
  float* new_adj = out + (long)Bb * Nn * Dd;          // [B,N,N,DE] = 16777216
  float* new_gf = new_adj + (long)Bb * Nn * Nn * DEe; // [B,DG] = 2048

  // workspace layout (floats)
  float* ws = (float*)d_ws;
  float* q_ws    = ws;                               // 262144
  float* k_ws    = q_ws + (long)Bb * Nn * Dd;        // 262144
  float* v_ws    = k_ws + (long)Bb * Nn * Dd;        // 262144
  float* attn_ws = v_ws + (long)Bb * Nn * Dd;        // 262144
  float* G1_ws   = attn_ws + (long)Bb * Nn * Dd;     // 2048
  float* G2_ws   = G1_ws + Bb * Dd;                  // 2048
  float* n2g_ws  = G2_ws + Bb * Dd;                  // 8192
  float* e2g_ws  = n2g_ws + Bb * 4 * Dd;             // 4096
  float* gsum_ws = e2g_ws + Bb * 4 * DEe;            // 2048

  const int M = Bb * Nn;                        // 1024 rows
  const float scaling = 0.17677669529663687f;   // 32^-0.5

  // QKV projections (node-masked; q also scaled)
  gemm_wmma_f32<<<128, 256, 0, stream>>>(query, W_q, b_q, node_mask, scaling, q_ws, M);
  gemm_wmma_f32<<<128, 256, 0, stream>>>(query, W_k, b_k, node_mask, 1.0f, k_ws, M);
  gemm_wmma_f32<<<128, 256, 0, stream>>>(query, W_v, b_v, node_mask, 1.0f, v_ws, M);

  // graph modulation vectors
  graph_mod<<<8, 256, 0, stream>>>(gfeat, W_gm, b_gm, W_ga, b_ga, G1_ws, G2_ws);

  // fused edge + attention core: one block per (b,t)
  const size_t shmem = (size_t)(Nn * DEe + Nn * Dd + 3 * Dd + Nn * Hh) * sizeof(float);
  edge_attn_fused<<<Bb * Nn, 256, shmem, stream>>>(
      q_ws, k_ws, v_ws, adj, W_em, b_em, W_ea, b_ea, W_eo, b_eo,
      G1_ws, G2_ws, attn_bias, edge_mask, new_adj, attn_ws);

  // graph-feature branch
  n2g_stats<<<8, 256, 0, stream>>>(query, n2g_ws);
  e2g_stats<<<4, 256, 0, stream>>>(adj, e2g_ws);
  graph_gather<<<8, 256, 0, stream>>>(n2g_ws, W_n2g, b_n2g, e2g_ws, W_e2g, b_e2g,
                                      gfeat, gsum_ws);
  graph_out_k<<<8, 256, 0, stream>>>(gsum_ws, W_go, b_go, new_gf);

  // output projection
  gemm_wmma_f32<<<128, 256, 0, stream>>>(attn_ws, W_o, b_o, nullptr, 1.0f, out, M);
}